// TensoRF_60679297958355
// MI455X (gfx1250) — compile-verified
//
#include <hip/hip_runtime.h>

// TensoRF VM-decomposition sampling for MI455X (gfx1250, wave32).
//
// Roofline: output stores 302 MB (compulsory) + planes 150 MB (fits the
// 192 MB L2; read from HBM once) + xyz 6 MB  =>  ~458 MB HBM floor ~ 20 us
// at 23.3 TB/s. The kernel is gather-latency / L2-request bound past that,
// so the design minimizes vmem issues (b64 row-pair gathers), keeps the
// plane set L2-resident (NT stores for output, NT load for xyz), and hides
// L2 latency (global_prefetch_b8 at distance-2 channels, unroll-2 MLP,
// light VGPR use for high wave32 occupancy). No matrix structure -> WMMA
// is not applicable to this workload.

typedef float f32x2 __attribute__((ext_vector_type(2), aligned(4)));
typedef float f32x3 __attribute__((ext_vector_type(3), aligned(4)));

#define GRES 512
#define NCH  48

__global__ __launch_bounds__(256) void tensorf_sample_kernel(
    const float* __restrict__ xyz,
    const float* __restrict__ p0,
    const float* __restrict__ p1,
    const float* __restrict__ p2,
    const float* __restrict__ l0,
    const float* __restrict__ l1,
    const float* __restrict__ l2,
    float* __restrict__ out,
    int N)
{
    const int n = blockIdx.x * blockDim.x + threadIdx.x;
    if (n >= N) return;

    // xyz is streamed exactly once: non-temporal b96 load, don't pollute L2.
    const f32x3 pt = __builtin_nontemporal_load(
        (const f32x3*)(xyz + 3 * (size_t)n));
    const float x = pt.x, y = pt.y, z = pt.z;

    // MAT_MODE = [(0,1),(0,2),(1,2)]  (gx -> W axis, gy -> H axis; plane[c, iy, ix])
    // VEC_MODE = [2,1,0]
    const float gx[3] = {x, x, y};
    const float gy[3] = {y, z, z};
    const float gz[3] = {z, y, x};

    const float scale = 0.5f * (float)(GRES - 1);

    int   poff[3], loff[3];
    float w00[3], w01[3], w10[3], w11[3], wz[3];

#pragma unroll
    for (int i = 0; i < 3; ++i) {
        float ix = (gx[i] + 1.0f) * scale;
        float iy = (gy[i] + 1.0f) * scale;
        float iz = (gz[i] + 1.0f) * scale;
        ix = fminf(fmaxf(ix, 0.0f), (float)(GRES - 1));
        iy = fminf(fmaxf(iy, 0.0f), (float)(GRES - 1));
        iz = fminf(fmaxf(iz, 0.0f), (float)(GRES - 1));
        // floor (inputs >= 0 after clamp); clamp low corner to G-2 so the
        // high corner is always +1 (border case becomes w == 1.0, same value).
        int ix0 = (int)ix; if (ix0 > GRES - 2) ix0 = GRES - 2;
        int iy0 = (int)iy; if (iy0 > GRES - 2) iy0 = GRES - 2;
        int iz0 = (int)iz; if (iz0 > GRES - 2) iz0 = GRES - 2;
        const float wx = ix - (float)ix0;
        const float wy = iy - (float)iy0;
        wz[i]   = iz - (float)iz0;
        // Bilinear corner weights (4-term FMA dot in the hot loop).
        const float w11v = wx * wy;
        w11[i] = w11v;
        w01[i] = wx - w11v;
        w10[i] = wy - w11v;
        w00[i] = 1.0f - wx - wy + w11v;
        poff[i] = iy0 * GRES + ix0;
        loff[i] = iz0;
    }

    const float* pA = p0 + poff[0];
    const float* pB = p1 + poff[1];
    const float* pC = p2 + poff[2];
    const float* lA = l0 + loff[0];
    const float* lB = l1 + loff[1];
    const float* lC = l2 + loff[2];

    const size_t CN = (size_t)NCH * (size_t)N;
    float* o0 = out + (size_t)n;
    float* o1 = o0 + CN;
    float* o2 = o1 + CN;

#pragma unroll 2
    for (int c = 0; c < NCH; ++c) {
        const size_t pc = (size_t)c * (GRES * GRES);
        const size_t lc = (size_t)c * GRES;

        // Each bilinear row is two adjacent floats -> one dword-aligned b64 gather.
        f32x2 a_top = *(const f32x2*)(pA + pc);
        f32x2 a_bot = *(const f32x2*)(pA + pc + GRES);
        f32x2 b_top = *(const f32x2*)(pB + pc);
        f32x2 b_bot = *(const f32x2*)(pB + pc + GRES);
        f32x2 c_top = *(const f32x2*)(pC + pc);
        f32x2 c_bot = *(const f32x2*)(pC + pc + GRES);
        f32x2 la    = *(const f32x2*)(lA + lc);
        f32x2 lb    = *(const f32x2*)(lB + lc);
        f32x2 lcv   = *(const f32x2*)(lC + lc);

        // Branchless distance-2 prefetch of future channel rows (+2 MB stride,
        // L2-resident) into L0; clamped so the address never leaves the tensor.
        {
            const int cnext = (c + 2 < NCH) ? (c + 2) : c;
            const size_t pn = (size_t)cnext * (GRES * GRES);
            __builtin_prefetch(pA + pn, 0, 3);
            __builtin_prefetch(pB + pn, 0, 3);
            __builtin_prefetch(pC + pn, 0, 3);
        }

        float av = a_top.x * w00[0];
        av = fmaf(a_top.y, w01[0], av);
        av = fmaf(a_bot.x, w10[0], av);
        av = fmaf(a_bot.y, w11[0], av);

        float bv = b_top.x * w00[1];
        bv = fmaf(b_top.y, w01[1], bv);
        bv = fmaf(b_bot.x, w10[1], bv);
        bv = fmaf(b_bot.y, w11[1], bv);

        float cv = c_top.x * w00[2];
        cv = fmaf(c_top.y, w01[2], cv);
        cv = fmaf(c_bot.x, w10[2], cv);
        cv = fmaf(c_bot.y, w11[2], cv);

        const float lav = fmaf(wz[0], la.y  - la.x,  la.x);
        const float lbv = fmaf(wz[1], lb.y  - lb.x,  lb.x);
        const float lcf = fmaf(wz[2], lcv.y - lcv.x, lcv.x);

        // Non-temporal streaming stores: keep the 150 MB plane set hot in L2.
        __builtin_nontemporal_store(av * lav, o0 + (size_t)c * N);
        __builtin_nontemporal_store(bv * lbv, o1 + (size_t)c * N);
        __builtin_nontemporal_store(cv * lcf, o2 + (size_t)c * N);
    }
}

extern "C" void kernel_launch(void* const* d_in, const int* in_sizes, int n_in,
                              void* d_out, int out_size, void* d_ws, size_t ws_size,
                              hipStream_t stream) {
    const float* xyz = (const float*)d_in[0];
    const float* p0  = (const float*)d_in[1];
    const float* p1  = (const float*)d_in[2];
    const float* p2  = (const float*)d_in[3];
    const float* l0  = (const float*)d_in[4];
    const float* l1  = (const float*)d_in[5];
    const float* l2  = (const float*)d_in[6];
    float* out = (float*)d_out;

    const int N = in_sizes[0] / 3;
    const int threads = 256;                 // 8 wave32 waves per block
    const int blocks  = (N + threads - 1) / threads;
    tensorf_sample_kernel<<<blocks, threads, 0, stream>>>(
        xyz, p0, p1, p2, l0, l1, l2, out, N);
}